// EncodingLayer_43473658970248
// MI455X (gfx1250) — compile-verified
//
#include <hip/hip_runtime.h>
#include <cstdint>

#define B_   32
#define S_   128
#define I_   64
#define H_   1024
#define T_   4

typedef float    f4    __attribute__((ext_vector_type(4)));
typedef uint32_t u32x4 __attribute__((ext_vector_type(4)));
typedef uint32_t u32x8 __attribute__((ext_vector_type(8)));

// ---------------------------------------------------------------------------
// Phase 1: drive[b,s,h] = sum_i x[b,s,i,h] * enc[i,h]
// grid = (H/256 h-chunks, 4096/32 bs-groups), block = 256 (8 waves).
// Encoding tile (64 rows x 256 floats = 64KB) staged to LDS by the TDM.
// ---------------------------------------------------------------------------
__global__ __launch_bounds__(256) void drive_kernel(const float* __restrict__ x,
                                                    const float* __restrict__ enc,
                                                    float* __restrict__ drive) {
  __shared__ f4 enc_s[I_ * 64];            // 64 rows x 64 f4 (256 floats) = 64 KB

  const uint32_t hc = blockIdx.x;          // h-chunk: 0..3, each 256 floats of H

  // ---- TDM: tensor_load_to_lds, 2D tensor (dim0=H elems, dim1=I rows),
  //      tile 256x64 starting at enc[0, hc*256], row stride H. One issue/block.
  if (threadIdx.x < 32u) {
    const uint64_t ga      = (uint64_t)(uintptr_t)(enc + (size_t)hc * 256u);
    const uint32_t lds_off = (uint32_t)(uintptr_t)(&enc_s[0]);  // low dword == LDS byte offset
    u32x4 g0;
    g0.x = 1u;                                                  // count=1 (valid), user mode
    g0.y = lds_off;                                             // lds_addr [63:32]
    g0.z = (uint32_t)ga;                                        // global_addr [95:64]
    g0.w = ((uint32_t)(ga >> 32) & 0x01FFFFFFu) | (2u << 30);   // addr[56:32] | type=2
    u32x8 g1;
    g1.s0 = 2u << 16;                                           // wg_mask=0, data_size=4B
    g1.s1 = ((uint32_t)H_ & 0xFFFFu) << 16;                     // abar=0 | tensor_dim0 lo16
    g1.s2 = ((uint32_t)H_ >> 16) | (((uint32_t)I_ & 0xFFFFu) << 16); // dim0 hi | dim1 lo
    g1.s3 = ((uint32_t)I_ >> 16) | (256u << 16);                // dim1 hi | tile_dim0=256
    g1.s4 = 64u;                                                // tile_dim1=64, tile_dim2=0
    g1.s5 = (uint32_t)H_;                                       // tensor_dim0_stride lo32
    g1.s6 = 0u;                                                 // stride0 hi | stride1 lo
    g1.s7 = 0u;                                                 // stride1 hi
    asm volatile("tensor_load_to_lds %0, %1" :: "s"(g0), "s"(g1) : "memory");
  }
  __builtin_amdgcn_s_wait_tensorcnt(0);
  __syncthreads();

  const uint32_t tid   = threadIdx.x;
  const uint32_t hl4   = tid & 63u;          // f4 lane within chunk (0..63)
  const uint32_t row   = tid >> 6;           // 0..3
  const uint32_t hidx4 = hc * 64u + hl4;     // f4 column in H/4 = 256
  const f4* x4 = (const f4*)x;
  f4*       d4 = (f4*)drive;

  for (uint32_t it = 0; it < 8u; ++it) {
    const uint32_t bs = blockIdx.y * 32u + it * 4u + row;      // 0..4095
    const f4* xs = x4 + (size_t)bs * I_ * (H_ / 4) + hidx4;
    f4 acc = {0.f, 0.f, 0.f, 0.f};
#pragma unroll 8
    for (uint32_t i = 0; i < I_; ++i) {
      f4 xv = __builtin_nontemporal_load(xs + (size_t)i * (H_ / 4));  // streamed, NT
      f4 ev = enc_s[i * 64u + hl4];                                   // ds_load_b128
      acc += xv * ev;                                                 // contracts to v_fma
    }
    d4[(size_t)bs * (H_ / 4) + hidx4] = acc;   // RT store -> stays in 192MB L2 for phase 2
  }
}

// ---------------------------------------------------------------------------
// Phase 2: sequential LIF scan, one thread per (b,h).
// grid = (H/64, B), block = 64 (2 waves). Reads drive (L2-hot), NT-stores spikes.
// ---------------------------------------------------------------------------
__global__ __launch_bounds__(64) void lif_kernel(const float* __restrict__ drive,
                                                 float* __restrict__ out) {
  const uint32_t h = blockIdx.x * 64u + threadIdx.x;
  const uint32_t b = blockIdx.y;
  const float* dp = drive + (size_t)b * S_ * H_ + h;
  float*       op = out   + (size_t)b * (S_ * T_) * H_ + h;

  float v = 0.0f;
  float d = dp[0];
  for (int s = 0; s < S_; ++s) {
    float dn = (s + 1 < S_) ? dp[(size_t)(s + 1) * H_] : 0.0f;  // load-ahead
#pragma unroll
    for (int t = 0; t < T_; ++t) {
      v = fmaf(0.9f, v, d);                          // v_new = DECAY*v + drive
      float e   = __expf(5.0f * (1.0f - v));         // exp(-BETA*(v-1)) via v_exp_f32
      float spk = __builtin_amdgcn_rcpf(1.0f + e);   // sigmoid; inf-safe at both ends
      v -= spk;                                      // subtractive reset (THRESHOLD=1)
      __builtin_nontemporal_store(spk, op + (size_t)(s * T_ + t) * H_);
    }
    d = dn;
  }
  // v_final, concatenated after output_sequence
  out[(size_t)B_ * (S_ * T_) * H_ + (size_t)b * H_ + h] = v;
}

// ---------------------------------------------------------------------------
extern "C" void kernel_launch(void* const* d_in, const int* in_sizes, int n_in,
                              void* d_out, int out_size, void* d_ws, size_t ws_size,
                              hipStream_t stream) {
  const float* x   = (const float*)d_in[0];   // [32,128,64,1024] f32
  const float* enc = (const float*)d_in[1];   // [64,1024] f32
  float* out   = (float*)d_out;               // 32*512*1024 spikes + 32*1024 v_final
  float* drive = (float*)d_ws;                // needs 32*128*1024*4 = 16 MiB scratch

  drive_kernel<<<dim3(H_ / 256, (B_ * S_) / 32), dim3(256), 0, stream>>>(x, enc, drive);
  lif_kernel  <<<dim3(H_ / 64, B_),              dim3(64),  0, stream>>>(drive, out);
}